// MGN_Net_37623913513587
// MI455X (gfx1250) — compile-verified
//
#include <hip/hip_runtime.h>
#include <hip/hip_bf16.h>
#include <math.h>

typedef float v2f __attribute__((ext_vector_type(2)));
typedef float v8f __attribute__((ext_vector_type(8)));

// ---------------------------------------------------------------------------
// Register-blocked fp32 WMMA GEMM: one wave computes an MT x NT grid of 16x16
// tiles (A fragments reused NT times, B fragments reused MT times).
//   C[b] (M x N) = A[b] (M x K) * B[b] (K x N) [+ Cinit] [+ bias_n] [relu]
// AT: A stored K x M (column major).  BT: B stored N x K (B^T row major).
// Lane layout per CDNA5 ISA (V_WMMA_F32_16X16X4_F32):
//   A/B: lane l -> idx (l&15), VGPR v holds k = k0 + 2*(l>>4) + v
//   C/D: VGPR r -> m = r + 8*(l>>4), n = l&15
// ---------------------------------------------------------------------------
template<int MT, int NT, bool AT, bool BT, bool RELU>
__global__ __launch_bounds__(32)
void k_gemm(const float* __restrict__ A, long asb, int lda,
            const float* __restrict__ Bm, long bsb, int ldb,
            float* __restrict__ C, long csb, int ldc,
            const float* __restrict__ bias,          // per-N bias or nullptr
            const float* __restrict__ Cinit, long cisb, int ldci, // or nullptr
            int K, int tilesM, int tilesN)
{
    int bid = blockIdx.x;
    int tn = bid % tilesN; bid /= tilesN;
    int tm = bid % tilesM; int b = bid / tilesM;

    const float* Ab = A  + (long)b * asb;
    const float* Bb = Bm + (long)b * bsb;

    int lane = threadIdx.x;
    int l16  = lane & 15;
    int hi   = lane >> 4;

    int am[MT], bn[NT];
#pragma unroll
    for (int i = 0; i < MT; ++i) am[i] = (tm * MT + i) * 16 + l16;
#pragma unroll
    for (int j = 0; j < NT; ++j) bn[j] = (tn * NT + j) * 16 + l16;

    v8f acc[MT][NT];
    if (Cinit) {
        const float* Ci = Cinit + (long)b * cisb;
#pragma unroll
        for (int i = 0; i < MT; ++i)
#pragma unroll
            for (int j = 0; j < NT; ++j)
#pragma unroll
                for (int r = 0; r < 8; ++r)
                    acc[i][j][r] =
                        Ci[(long)((tm * MT + i) * 16 + r + 8 * hi) * ldci + bn[j]];
    } else {
#pragma unroll
        for (int i = 0; i < MT; ++i)
#pragma unroll
            for (int j = 0; j < NT; ++j)
#pragma unroll
                for (int r = 0; r < 8; ++r) acc[i][j][r] = 0.0f;
    }

    for (int k0 = 0; k0 < K; k0 += 4) {
        int k = k0 + 2 * hi;
        v2f a[MT], bv[NT];
#pragma unroll
        for (int i = 0; i < MT; ++i) {
            if (AT) { a[i].x = Ab[(long)k * lda + am[i]];
                      a[i].y = Ab[(long)(k + 1) * lda + am[i]]; }
            else    { a[i].x = Ab[(long)am[i] * lda + k];
                      a[i].y = Ab[(long)am[i] * lda + k + 1];   }
        }
#pragma unroll
        for (int j = 0; j < NT; ++j) {
            if (BT) { bv[j].x = Bb[(long)bn[j] * ldb + k];
                      bv[j].y = Bb[(long)bn[j] * ldb + k + 1];  }
            else    { bv[j].x = Bb[(long)k * ldb + bn[j]];
                      bv[j].y = Bb[(long)(k + 1) * ldb + bn[j]];}
        }
#pragma unroll
        for (int i = 0; i < MT; ++i)
#pragma unroll
            for (int j = 0; j < NT; ++j)
                acc[i][j] = __builtin_amdgcn_wmma_f32_16x16x4_f32(
                    false, a[i], false, bv[j], (short)0, acc[i][j], false, false);
    }

    float* Cb = C + (long)b * csb;
#pragma unroll
    for (int i = 0; i < MT; ++i)
#pragma unroll
        for (int j = 0; j < NT; ++j)
#pragma unroll
            for (int r = 0; r < 8; ++r) {
                int m = (tm * MT + i) * 16 + r + 8 * hi;
                float v = acc[i][j][r];
                if (bias) v += bias[bn[j]];
                if (RELU) v = fmaxf(v, 0.0f);
                Cb[(long)m * ldc + bn[j]] = v;
            }
}

// ---------------------------------------------------------------------------
// Row softmax, one wave per row (in place).
// ---------------------------------------------------------------------------
__global__ __launch_bounds__(32)
void k_softmax(float* __restrict__ p, int cols, int ld)
{
    float* row = p + (long)blockIdx.x * ld;
    int lane = threadIdx.x;
    float m = -INFINITY;
    for (int c = lane; c < cols; c += 32) m = fmaxf(m, row[c]);
    for (int o = 16; o > 0; o >>= 1) m = fmaxf(m, __shfl_xor(m, o, 32));
    float s = 0.0f;
    for (int c = lane; c < cols; c += 32) {
        float e = __expf(row[c] - m);
        row[c] = e; s += e;
    }
    for (int o = 16; o > 0; o >>= 1) s += __shfl_xor(s, o, 32);
    float inv = 1.0f / s;
    for (int c = lane; c < cols; c += 32) row[c] *= inv;
}

// ---------------------------------------------------------------------------
// modal_1[b,d] = mean_t x[b,d,t]; modal_2[b,d] = mean_i vis_fc[b,i,d]
// ---------------------------------------------------------------------------
__global__ void k_modal(const float* __restrict__ x,
                        const float* __restrict__ visfc,
                        float* __restrict__ modal, int B)
{
    int idx = blockIdx.x * blockDim.x + threadIdx.x;
    if (idx >= B * 768) return;
    int b = idx / 768, d = idx % 768;
    const float* xr = x + ((long)b * 768 + d) * 512;
    float s = 0.0f;
    for (int t = 0; t < 512; ++t) s += xr[t];
    const float* vf = visfc + (long)b * 256 * 768 + d;
    float s2 = 0.0f;
    for (int i = 0; i < 256; ++i) s2 += vf[(long)i * 768];
    modal[(long)b * 1536 + d]       = s / 512.0f;
    modal[(long)b * 1536 + 768 + d] = s2 / 256.0f;
}

// ---------------------------------------------------------------------------
// Router MLP: 1536 -> 128 (relu) -> 32 (relu) -> 4 -> softmax.  One block/batch.
// ---------------------------------------------------------------------------
__global__ __launch_bounds__(128)
void k_router(const float* __restrict__ modal,
              const float* __restrict__ r1w, const float* __restrict__ r1b,
              const float* __restrict__ r2w, const float* __restrict__ r2b,
              const float* __restrict__ r3w, const float* __restrict__ r3b,
              float* __restrict__ probs)
{
    __shared__ float h1[128];
    __shared__ float h2[32];
    __shared__ float lg[4];
    int b = blockIdx.x, t = threadIdx.x;
    const float* h = modal + (long)b * 1536;

    float s = r1b[t];
    const float* w = r1w + (long)t * 1536;
    for (int i = 0; i < 1536; ++i) s += w[i] * h[i];
    h1[t] = fmaxf(s, 0.0f);
    __syncthreads();

    if (t < 32) {
        float s2 = r2b[t];
        const float* w2 = r2w + t * 128;
        for (int i = 0; i < 128; ++i) s2 += w2[i] * h1[i];
        h2[t] = fmaxf(s2, 0.0f);
    }
    __syncthreads();

    if (t < 4) {
        float s3 = r3b[t];
        const float* w3 = r3w + t * 32;
        for (int i = 0; i < 32; ++i) s3 += w3[i] * h2[i];
        lg[t] = s3;
    }
    __syncthreads();

    if (t == 0) {
        float mx = fmaxf(fmaxf(lg[0], lg[1]), fmaxf(lg[2], lg[3]));
        float e0 = __expf(lg[0] - mx), e1 = __expf(lg[1] - mx);
        float e2 = __expf(lg[2] - mx), e3 = __expf(lg[3] - mx);
        float inv = 1.0f / (e0 + e1 + e2 + e3);
        probs[b * 4 + 0] = e0 * inv; probs[b * 4 + 1] = e1 * inv;
        probs[b * 4 + 2] = e2 * inv; probs[b * 4 + 3] = e3 * inv;
    }
}

// lb_loss: pm.shape[0]==1 so u==1.0 -> loss = -sum_e log(mean_b probs[b,e])
__global__ void k_lb(const float* __restrict__ probs, float* __restrict__ out, int B)
{
    if (threadIdx.x == 0 && blockIdx.x == 0) {
        float l = 0.0f;
        for (int e = 0; e < 4; ++e) {
            float pm = 0.0f;
            for (int b = 0; b < B; ++b) pm += probs[b * 4 + e];
            pm /= (float)B;
            l -= logf(pm);
        }
        out[0] = l;
    }
}

// ---------------------------------------------------------------------------
// h[t,d] = LN_d( x[b,d,t] + gate_av * xres[t,d] ) * w[d] + bias[d]
// ---------------------------------------------------------------------------
__global__ __launch_bounds__(256)
void k_ln_before(const float* __restrict__ x, const float* __restrict__ xres,
                 const float* __restrict__ gate_av,
                 const float* __restrict__ w, const float* __restrict__ bia,
                 float* __restrict__ h)
{
    int row = blockIdx.x;                 // b*512 + t
    int b = row >> 9, t = row & 511;
    const float* xr = x + ((long)b * 768) * 512 + t;
    const float* rr = xres + (long)row * 768;
    float g = gate_av[0];

    __shared__ float red[256];
    float vals[3];
    float s = 0.0f;
#pragma unroll
    for (int j = 0; j < 3; ++j) {
        int d = threadIdx.x + j * 256;
        float v = xr[(long)d * 512] + g * rr[d];
        vals[j] = v; s += v;
    }
    red[threadIdx.x] = s; __syncthreads();
    for (int o = 128; o > 0; o >>= 1) {
        if (threadIdx.x < o) red[threadIdx.x] += red[threadIdx.x + o];
        __syncthreads();
    }
    float mu = red[0] / 768.0f; __syncthreads();
    float vs = 0.0f;
#pragma unroll
    for (int j = 0; j < 3; ++j) { float d0 = vals[j] - mu; vs += d0 * d0; }
    red[threadIdx.x] = vs; __syncthreads();
    for (int o = 128; o > 0; o >>= 1) {
        if (threadIdx.x < o) red[threadIdx.x] += red[threadIdx.x + o];
        __syncthreads();
    }
    float rstd = rsqrtf(red[0] / 768.0f + 1e-5f);
    float* hr = h + (long)row * 768;
#pragma unroll
    for (int j = 0; j < 3; ++j) {
        int d = threadIdx.x + j * 256;
        hr[d] = (vals[j] - mu) * rstd * w[d] + bia[d];
    }
}

// ---------------------------------------------------------------------------
// out[b,d,t] (+)= probs[b,e] * gate * ( LN_d(outpre[t,:]) * w + bias )
// ---------------------------------------------------------------------------
__global__ __launch_bounds__(256)
void k_ln_post_combine(const float* __restrict__ outpre,
                       const float* __restrict__ gate,
                       const float* __restrict__ w, const float* __restrict__ bia,
                       const float* __restrict__ probs, int eidx, int first,
                       float* __restrict__ out)
{
    int row = blockIdx.x;
    int b = row >> 9, t = row & 511;
    const float* orow = outpre + (long)row * 768;

    __shared__ float red[256];
    float vals[3];
    float s = 0.0f;
#pragma unroll
    for (int j = 0; j < 3; ++j) {
        int d = threadIdx.x + j * 256;
        float v = orow[d];
        vals[j] = v; s += v;
    }
    red[threadIdx.x] = s; __syncthreads();
    for (int o = 128; o > 0; o >>= 1) {
        if (threadIdx.x < o) red[threadIdx.x] += red[threadIdx.x + o];
        __syncthreads();
    }
    float mu = red[0] / 768.0f; __syncthreads();
    float vs = 0.0f;
#pragma unroll
    for (int j = 0; j < 3; ++j) { float d0 = vals[j] - mu; vs += d0 * d0; }
    red[threadIdx.x] = vs; __syncthreads();
    for (int o = 128; o > 0; o >>= 1) {
        if (threadIdx.x < o) red[threadIdx.x] += red[threadIdx.x + o];
        __syncthreads();
    }
    float rstd  = rsqrtf(red[0] / 768.0f + 1e-5f);
    float scale = gate[0] * probs[b * 4 + eidx];
    float* od = out + ((long)b * 768) * 512 + t;
#pragma unroll
    for (int j = 0; j < 3; ++j) {
        int d = threadIdx.x + j * 256;
        float v = ((vals[j] - mu) * rstd * w[d] + bia[d]) * scale;
        if (first) od[(long)d * 512] = v;
        else       od[(long)d * 512] += v;
    }
}

// ---------------------------------------------------------------------------
extern "C" void kernel_launch(void* const* d_in, const int* in_sizes, int n_in,
                              void* d_out, int out_size, void* d_ws, size_t ws_size,
                              hipStream_t stream)
{
    const int B = 16, D = 768, T = 512, Nv = 256, Cc = 1536, Nn = 144, Tk = 32, Dd = 96;
    (void)in_sizes; (void)n_in; (void)out_size; (void)ws_size;

    // JAX pytree flattening: dict keys sorted alphabetically, lists in order.
    const float* x      = (const float*)d_in[0];
    const float* vis    = (const float*)d_in[1];
    const float* conv_b = (const float*)d_in[2];
    const float* conv_w = (const float*)d_in[3];
    const float* fc_b   = (const float*)d_in[4];
    const float* fc_w   = (const float*)d_in[5];
    const float* r1_b   = (const float*)d_in[24];
    const float* r1_w   = (const float*)d_in[25];
    const float* r2_b   = (const float*)d_in[26];
    const float* r2_w   = (const float*)d_in[27];
    const float* r3_b   = (const float*)d_in[28];
    const float* r3_w   = (const float*)d_in[29];

    float* ws      = (float*)d_ws;
    float* vt_t    = ws;                               // [B,144,256]
    float* visfc   = vt_t  + (long)B * Nn * Nv;        // [B,256,768]
    float* modal   = visfc + (long)B * Nv * D;         // [B,1536]
    float* probs   = modal + (long)B * 1536;           // [B,4]
    float* scratch = probs + B * 4;                    // union (mm: rep/s1/s2, sm: s3)
    float* rep     = scratch;                          // [B,32,768]
    float* s1      = rep + (long)B * Tk * D;           // [B,32,256]
    float* s2      = s1  + (long)B * Tk * Nv;          // [B,512,32]
    float* s3      = scratch;                          // [B,512,512]
    float* xres    = scratch + (long)B * T * T;        // [B,512,768]
    float* h       = xres + (long)B * T * D;           // [B,512,768]
    float* z       = h    + (long)B * T * D;           // [B,512,96]
    float* outp    = xres;                             // alias: xres dead after ln_before

    float* fin = (float*)d_out;

    // --- vision adapter ---
    // vt_t[b,n,o] = sum_c vis[b,n,c]*conv_w[o,c] + conv_b[o]  (M=144,N=256,K=1536)
    k_gemm<3, 4, false, true, false><<<B * 3 * 4, 32, 0, stream>>>(
        vis, (long)Nn * Cc, Cc, conv_w, 0, Cc, vt_t, (long)Nn * Nv, Nv,
        conv_b, nullptr, 0, 0, Cc, 3, 4);
    // visfc[b,i,d] = sum_n vt_t[b,n,i]*fc_w[d,n] + fc_b[d]    (M=256,N=768,K=144)
    k_gemm<4, 4, true, true, false><<<B * 4 * 12, 32, 0, stream>>>(
        vt_t, (long)Nn * Nv, Nv, fc_w, 0, Nn, visfc, (long)Nv * D, D,
        fc_b, nullptr, 0, 0, Nn, 4, 12);

    // --- router ---
    k_modal<<<(B * D + 255) / 256, 256, 0, stream>>>(x, visfc, modal, B);
    k_router<<<B, 128, 0, stream>>>(modal, r1_w, r1_b, r2_w, r2_b, r3_w, r3_b, probs);
    k_lb<<<1, 32, 0, stream>>>(probs, fin + (long)B * D * T, B);

    // --- experts ---
    for (int e = 0; e < 4; ++e) {
        bool mm = (e < 2);
        int pb = mm ? (6 + 9 * e) : (30 + 8 * (e - 2));
        const float* down_w = (const float*)d_in[pb + 0];
        const float* gate   = (const float*)d_in[pb + 1];
        const float* gateav = (const float*)d_in[pb + 2];
        const float* lnb_b  = (const float*)d_in[pb + 3];
        const float* lnb_w  = (const float*)d_in[pb + 4];
        const float* lnp_b  = (const float*)d_in[pb + 5];
        const float* lnp_w  = (const float*)d_in[pb + 6];
        const float* my_tok = mm ? (const float*)d_in[pb + 7] : nullptr;
        const float* up_w   = (const float*)d_in[pb + (mm ? 8 : 7)];

        if (mm) {
            // s1[b,k,i] = my_tok[k,:] . visfc[b,i,:]           (M=32,N=256,K=768)
            k_gemm<2, 4, false, true, false><<<B * 1 * 4, 32, 0, stream>>>(
                my_tok, 0, D, visfc, (long)Nv * D, D, s1, (long)Tk * Nv, Nv,
                nullptr, nullptr, 0, 0, D, 1, 4);
            k_softmax<<<B * Tk, 32, 0, stream>>>(s1, Nv, Nv);
            // rep = my_tok + s1 * visfc                         (M=32,N=768,K=256)
            k_gemm<2, 4, false, false, false><<<B * 1 * 12, 32, 0, stream>>>(
                s1, (long)Tk * Nv, Nv, visfc, (long)Nv * D, D, rep, (long)Tk * D, D,
                nullptr, my_tok, 0, D, Nv, 1, 12);
            // s2[b,t,k] = x[b,:,t] . rep[k,:]                   (M=512,N=32,K=768)
            k_gemm<4, 2, true, true, false><<<B * 8 * 1, 32, 0, stream>>>(
                x, (long)D * T, T, rep, (long)Tk * D, D, s2, (long)T * Tk, Tk,
                nullptr, nullptr, 0, 0, D, 8, 1);
            k_softmax<<<B * T, 32, 0, stream>>>(s2, Tk, Tk);
            // xres[b,t,d] = s2 * rep                            (M=512,N=768,K=32)
            k_gemm<4, 4, false, false, false><<<B * 8 * 12, 32, 0, stream>>>(
                s2, (long)T * Tk, Tk, rep, (long)Tk * D, D, xres, (long)T * D, D,
                nullptr, nullptr, 0, 0, Tk, 8, 12);
        } else {
            // s3[b,t,s] = x[b,:,t] . x[b,:,s]                   (M=512,N=512,K=768)
            k_gemm<4, 4, true, false, false><<<B * 8 * 8, 32, 0, stream>>>(
                x, (long)D * T, T, x, (long)D * T, T, s3, (long)T * T, T,
                nullptr, nullptr, 0, 0, D, 8, 8);
            k_softmax<<<B * T, 32, 0, stream>>>(s3, T, T);
            // xres[b,s,d] = sum_t att[t,s] * x[b,d,t]           (M=512,N=768,K=512)
            k_gemm<4, 4, true, true, false><<<B * 8 * 12, 32, 0, stream>>>(
                s3, (long)T * T, T, x, (long)D * T, T, xres, (long)T * D, D,
                nullptr, nullptr, 0, 0, T, 8, 12);
        }

        k_ln_before<<<B * T, 256, 0, stream>>>(x, xres, gateav, lnb_w, lnb_b, h);

        // grouped 1x1 down conv (groups=2); relu only in multimodal path
        for (int g = 0; g < 2; ++g) {
            if (mm)
                k_gemm<4, 3, false, true, true><<<B * 8 * 1, 32, 0, stream>>>(
                    h + g * 384, (long)T * D, D, down_w + (long)g * 48 * 384, 0, 384,
                    z + g * 48, (long)T * Dd, Dd, nullptr, nullptr, 0, 0, 384, 8, 1);
            else
                k_gemm<4, 3, false, true, false><<<B * 8 * 1, 32, 0, stream>>>(
                    h + g * 384, (long)T * D, D, down_w + (long)g * 48 * 384, 0, 384,
                    z + g * 48, (long)T * Dd, Dd, nullptr, nullptr, 0, 0, 384, 8, 1);
        }
        // grouped 1x1 up conv
        for (int g = 0; g < 2; ++g) {
            k_gemm<4, 4, false, true, false><<<B * 8 * 6, 32, 0, stream>>>(
                z + g * 48, (long)T * Dd, Dd, up_w + (long)g * 384 * 48, 0, 48,
                outp + g * 384, (long)T * D, D, nullptr, nullptr, 0, 0, 48, 8, 6);
        }

        k_ln_post_combine<<<B * T, 256, 0, stream>>>(
            outp, gate, lnp_w, lnp_b, probs, e, (e == 0) ? 1 : 0, fin);
    }
}